// KPlaneEmbedder_77489799954576
// MI455X (gfx1250) — compile-verified
//
#include <hip/hip_runtime.h>
#include <stdint.h>
#include <stddef.h>

#define DEVINL __device__ __forceinline__

typedef __attribute__((ext_vector_type(4))) float f4;

namespace {

constexpr int RES0 = 64, RES1 = 128, RES2 = 256, RES3 = 512;
constexpr size_t PLANE_F0 = (size_t)RES0 * RES0 * 8;   // floats per transposed plane
constexpr size_t PLANE_F1 = (size_t)RES1 * RES1 * 8;
constexpr size_t PLANE_F2 = (size_t)RES2 * RES2 * 8;
constexpr size_t PLANE_F3 = (size_t)RES3 * RES3 * 8;
constexpr size_t LBASE0 = 0;
constexpr size_t LBASE1 = LBASE0 + 6 * PLANE_F0;
constexpr size_t LBASE2 = LBASE1 + 6 * PLANE_F1;
constexpr size_t LBASE3 = LBASE2 + 6 * PLANE_F2;
constexpr size_t TOTAL_F = LBASE3 + 6 * PLANE_F3;      // 16,711,680 floats = 66.8 MB

constexpr int CI[6] = {0, 0, 0, 1, 1, 2};
constexpr int CJ[6] = {1, 2, 3, 2, 3, 3};

// Per-level coordinate prep: t = (x+1)*0.5 precomputed once per point.
template <int R>
DEVINL void coord_prep(float t, int& i0, int& i1, float& w) {
  constexpr float RM1 = (float)(R - 1);
  float fx = fminf(fmaxf(t * RM1, 0.0f), RM1);
  float f0 = floorf(fx);
  w = fx - f0;
  i0 = (int)f0;
  i1 = min(i0 + 1, R - 1);
}

// ---------- transposed-layout level: 8 channels contiguous per texel ----------
template <int R>
DEVINL void level_feat_tr(const float* __restrict__ base, const f4& t, f4& lo, f4& hi) {
  constexpr size_t PS = (size_t)R * R * 8;
  // Hoisted per-coordinate prep (4 preps instead of 12 per level).
  int xo0[4], xo1[4], yo0[4], yo1[4];
  float w[4];
#pragma unroll
  for (int c = 0; c < 4; ++c) {
    int i0, i1; float wc;
    coord_prep<R>(t[c], i0, i1, wc);
    w[c]   = wc;
    xo0[c] = i0 * 8;        // float offset of column when used as x
    xo1[c] = i1 * 8;
    yo0[c] = i0 * (R * 8);  // float offset of row when used as y
    yo1[c] = i1 * (R * 8);
  }
  lo = (f4)1.0f;
  hi = (f4)1.0f;
#pragma unroll
  for (int k = 0; k < 6; ++k) {
    const int i = CI[k], j = CJ[k];
    const float wx = w[i], wy = w[j];
    const float omx = 1.0f - wx, omy = 1.0f - wy;
    const float w00 = omx * omy, w01 = wx * omy, w10 = omx * wy, w11 = wx * wy;
    const float* pl = base + (size_t)k * PS;
    const f4* p00 = reinterpret_cast<const f4*>(pl + (yo0[j] + xo0[i]));
    const f4* p01 = reinterpret_cast<const f4*>(pl + (yo0[j] + xo1[i]));
    const f4* p10 = reinterpret_cast<const f4*>(pl + (yo1[j] + xo0[i]));
    const f4* p11 = reinterpret_cast<const f4*>(pl + (yo1[j] + xo1[i]));
    f4 a0 = p00[0], b0 = p01[0], c0 = p10[0], d0 = p11[0];
    f4 a1 = p00[1], b1 = p01[1], c1 = p10[1], d1 = p11[1];
    f4 vlo = a0 * w00 + b0 * w01 + c0 * w10 + d0 * w11;  // fp-contract -> (pk) fma
    f4 vhi = a1 * w00 + b1 * w01 + c1 * w10 + d1 * w11;
    lo *= vlo;
    hi *= vhi;
  }
}

// ---------- strided-layout level (fallback, original (8,r,r) layout) ----------
template <int R>
DEVINL void level_feat_str(const float* const* __restrict__ planes, const f4& t, f4& lo, f4& hi) {
  constexpr size_t CS = (size_t)R * R;
  int x0[4], x1[4], y0[4], y1[4];
  float w[4];
#pragma unroll
  for (int c = 0; c < 4; ++c) {
    int i0, i1; float wc;
    coord_prep<R>(t[c], i0, i1, wc);
    w[c]  = wc;
    x0[c] = i0;      x1[c] = i1;
    y0[c] = i0 * R;  y1[c] = i1 * R;
  }
  lo = (f4)1.0f;
  hi = (f4)1.0f;
#pragma unroll
  for (int k = 0; k < 6; ++k) {
    const int i = CI[k], j = CJ[k];
    const float wx = w[i], wy = w[j];
    const float omx = 1.0f - wx, omy = 1.0f - wy;
    const float w00 = omx * omy, w01 = wx * omy, w10 = omx * wy, w11 = wx * wy;
    const float* pl = planes[k];
    size_t i00 = (size_t)(y0[j] + x0[i]), i01 = (size_t)(y0[j] + x1[i]);
    size_t i10 = (size_t)(y1[j] + x0[i]), i11 = (size_t)(y1[j] + x1[i]);
    f4 vlo, vhi;
#pragma unroll
    for (int c = 0; c < 4; ++c) {
      const float* b = pl + (size_t)c * CS;
      vlo[c] = b[i00] * w00 + b[i01] * w01 + b[i10] * w10 + b[i11] * w11;
    }
#pragma unroll
    for (int c = 0; c < 4; ++c) {
      const float* b = pl + (size_t)(c + 4) * CS;
      vhi[c] = b[i00] * w00 + b[i01] * w01 + b[i10] * w10 + b[i11] * w11;
    }
    lo *= vlo;
    hi *= vhi;
  }
}

} // namespace

// ---------- prep: repack one (8,r,r) grid into channel-interleaved (r,r,8) ----------
__global__ void kplanes_transpose(const float* __restrict__ src, float* __restrict__ dst, int r) {
  int t = blockIdx.x * blockDim.x + threadIdx.x;
  int n = r * r;
  if (t >= n) return;
  f4 lo, hi;
#pragma unroll
  for (int c = 0; c < 4; ++c) lo[c] = src[(size_t)c * n + t];
#pragma unroll
  for (int c = 0; c < 4; ++c) hi[c] = src[(size_t)(c + 4) * n + t];
  f4* d = reinterpret_cast<f4*>(dst + ((size_t)t << 3));
  d[0] = lo;   // regular-temporal stores: keep grids resident in the 192MB L2
  d[1] = hi;
}

// ---------- main kernel: async-LDS staged coords, L2-resident transposed gathers,
// ---------- NT streaming output ----------
__global__ void __launch_bounds__(256) kplanes_main(const float* __restrict__ pts,
                                                    const float* __restrict__ tg,
                                                    float* __restrict__ out, int npts) {
  __shared__ f4 sbuf[2][256];
  const int tid = threadIdx.x;
  const int ntiles = (npts + 255) >> 8;

  auto issue = [&](int b, int tile) {
    int idx = (tile << 8) + tid;
    if (idx >= npts) idx = npts - 1;
    unsigned lds_addr = (unsigned)(uintptr_t)(&sbuf[b][tid]);
    unsigned long long gaddr = (unsigned long long)(uintptr_t)(pts + ((size_t)idx << 2));
    // CDNA5 async copy: global -> LDS, tracked by ASYNCcnt
    asm volatile("global_load_async_to_lds_b128 %0, %1, off"
                 :: "v"(lds_addr), "v"(gaddr) : "memory");
  };

  int buf = 0;
  int tile = blockIdx.x;
  if (tile < ntiles) issue(0, tile);

  for (; tile < ntiles; tile += gridDim.x) {
    int nxt = tile + gridDim.x;
    if (nxt < ntiles) {
      issue(buf ^ 1, nxt);                               // prefetch next tile
      asm volatile("s_wait_asynccnt 0x1" ::: "memory");  // current tile done, next in flight
    } else {
      asm volatile("s_wait_asynccnt 0x0" ::: "memory");
    }
    __syncthreads();

    const int idx = (tile << 8) + tid;
    if (idx < npts) {
      f4 p = sbuf[buf][tid];
      f4 t = (p + 1.0f) * 0.5f;                          // shared across all levels
      float* o = out + ((size_t)idx << 5);
      f4 lo, hi;
      level_feat_tr<RES0>(tg + LBASE0, t, lo, hi);
      __builtin_nontemporal_store(lo, reinterpret_cast<f4*>(o + 0));
      __builtin_nontemporal_store(hi, reinterpret_cast<f4*>(o + 4));
      level_feat_tr<RES1>(tg + LBASE1, t, lo, hi);
      __builtin_nontemporal_store(lo, reinterpret_cast<f4*>(o + 8));
      __builtin_nontemporal_store(hi, reinterpret_cast<f4*>(o + 12));
      level_feat_tr<RES2>(tg + LBASE2, t, lo, hi);
      __builtin_nontemporal_store(lo, reinterpret_cast<f4*>(o + 16));
      __builtin_nontemporal_store(hi, reinterpret_cast<f4*>(o + 20));
      level_feat_tr<RES3>(tg + LBASE3, t, lo, hi);
      __builtin_nontemporal_store(lo, reinterpret_cast<f4*>(o + 24));
      __builtin_nontemporal_store(hi, reinterpret_cast<f4*>(o + 28));
    }
    __syncthreads();  // WAR guard before reusing this LDS buffer
    buf ^= 1;
  }
}

// ---------- fallback if workspace can't hold the transposed grids ----------
struct GridPtrs { const float* g[24]; };

__global__ void __launch_bounds__(256) kplanes_direct(const float* __restrict__ pts,
                                                      GridPtrs gp,
                                                      float* __restrict__ out, int npts) {
  int idx = blockIdx.x * 256 + threadIdx.x;
  const int stride = gridDim.x * 256;
  for (; idx < npts; idx += stride) {
    f4 p = *reinterpret_cast<const f4*>(pts + ((size_t)idx << 2));
    f4 t = (p + 1.0f) * 0.5f;
    float* o = out + ((size_t)idx << 5);
    f4 lo, hi;
    level_feat_str<RES0>(&gp.g[0], t, lo, hi);
    __builtin_nontemporal_store(lo, reinterpret_cast<f4*>(o + 0));
    __builtin_nontemporal_store(hi, reinterpret_cast<f4*>(o + 4));
    level_feat_str<RES1>(&gp.g[6], t, lo, hi);
    __builtin_nontemporal_store(lo, reinterpret_cast<f4*>(o + 8));
    __builtin_nontemporal_store(hi, reinterpret_cast<f4*>(o + 12));
    level_feat_str<RES2>(&gp.g[12], t, lo, hi);
    __builtin_nontemporal_store(lo, reinterpret_cast<f4*>(o + 16));
    __builtin_nontemporal_store(hi, reinterpret_cast<f4*>(o + 20));
    level_feat_str<RES3>(&gp.g[18], t, lo, hi);
    __builtin_nontemporal_store(lo, reinterpret_cast<f4*>(o + 24));
    __builtin_nontemporal_store(hi, reinterpret_cast<f4*>(o + 28));
  }
}

extern "C" void kernel_launch(void* const* d_in, const int* in_sizes, int n_in,
                              void* d_out, int out_size, void* d_ws, size_t ws_size,
                              hipStream_t stream) {
  (void)n_in; (void)out_size;
  const float* pts = (const float*)d_in[0];
  float* out = (float*)d_out;
  const int npts = in_sizes[0] / 4;

  static const int    RESL[4]  = {64, 128, 256, 512};
  static const size_t LBASE[4] = {LBASE0, LBASE1, LBASE2, LBASE3};

  if (ws_size >= TOTAL_F * sizeof(float)) {
    float* tg = (float*)d_ws;
    int k = 1;
    for (int l = 0; l < 4; ++l) {
      const int r = RESL[l];
      const int n = r * r;
      const size_t pf = (size_t)n * 8;
      for (int pl = 0; pl < 6; ++pl) {
        kplanes_transpose<<<(n + 255) / 256, 256, 0, stream>>>(
            (const float*)d_in[k], tg + LBASE[l] + (size_t)pl * pf, r);
        ++k;
      }
    }
    kplanes_main<<<2048, 256, 0, stream>>>(pts, tg, out, npts);
  } else {
    GridPtrs gp;
    for (int i = 0; i < 24; ++i) gp.g[i] = (const float*)d_in[1 + i];
    kplanes_direct<<<2048, 256, 0, stream>>>(pts, gp, out, npts);
  }
}